// ImprovedGNN_91345364451329
// MI455X (gfx1250) — compile-verified
//
#include <hip/hip_runtime.h>

// ---------------------------------------------------------------------------
// GCN (3x GCNConv + BN + ReLU) + mean-pool + MLP head for MI455X (gfx1250).
// Bandwidth-bound (~4 GB traffic / 23.3 TB/s); GEMMs (1.6 GFLOP) done exactly
// in f32 via V_WMMA_F32_16X16X4_F32 (one 16-node tile per wave32).
// ---------------------------------------------------------------------------

namespace {
constexpr int   kN     = 100000;    // nodes
constexpr int   kE     = 3200000;   // edges
constexpr int   kF     = 128;       // input features
constexpr int   kG     = 64;        // graphs
constexpr float kEps   = 1e-5f;
constexpr int   kTiles = kN / 16;   // 6250 full 16-row tiles (kN % 16 == 0)
}

typedef float v2f __attribute__((ext_vector_type(2)));
typedef float v8f __attribute__((ext_vector_type(8)));

// ------------------------------ degrees ------------------------------------
__global__ void k_deg_init(float* __restrict__ deg) {
  int i = blockIdx.x * blockDim.x + threadIdx.x;
  if (i < kN) deg[i] = 1.0f;  // self-loop
}

__global__ void k_deg_acc(const int* __restrict__ dst, float* __restrict__ deg) {
  int e = blockIdx.x * blockDim.x + threadIdx.x;
  if (e < kE) atomicAdd(&deg[dst[e]], 1.0f);
}

__global__ void k_dinv(const float* __restrict__ deg, float* __restrict__ dinv) {
  int i = blockIdx.x * blockDim.x + threadIdx.x;
  if (i < kN) dinv[i] = rsqrtf(deg[i]);
}

// ------------------------------ WMMA GEMM ----------------------------------
// out[kN, Fo] = h[kN, Fi] @ W[Fi, Fo]     (Fi % 4 == 0, Fo % 16 == 0)
// One wave computes a 16-row tile across all Fo columns using
// V_WMMA_F32_16X16X4_F32 (A: 16x4 f32, B: 4x16 f32, C/D: 16x16 f32).
template <int Fi, int Fo>
__global__ void k_gemm_wmma(const float* __restrict__ h,
                            const float* __restrict__ W,
                            float* __restrict__ out) {
  const int wave = threadIdx.x >> 5;
  const int lane = threadIdx.x & 31;
  const int tile = blockIdx.x * (blockDim.x >> 5) + wave;
  if (tile >= kTiles) return;                 // wave-uniform: EXEC stays all-1s
  const int row0 = tile << 4;
  const int lrow = lane & 15;                 // M (for A) / N (for B,C,D)
  const int half = lane >> 4;                 // selects K-pair / row group

  const float* arow = h + (size_t)(row0 + lrow) * Fi + 2 * half;

  v8f acc[Fo / 16];
  const v8f vzero = {};
#pragma unroll
  for (int t = 0; t < Fo / 16; ++t) acc[t] = vzero;

#pragma unroll 4
  for (int k = 0; k < Fi; k += 4) {
    // A fragment: row = row0+lrow, K = k + 2*half + {0,1}  (8B-aligned pair)
    v2f a = *(const v2f*)(arow + k);
    const int ka = k + 2 * half;
#pragma unroll
    for (int t = 0; t < Fo / 16; ++t) {
      const int ncol = t * 16 + lrow;
      v2f b;
      b.x = W[(size_t)ka * Fo + ncol];
      b.y = W[(size_t)(ka + 1) * Fo + ncol];
      acc[t] = __builtin_amdgcn_wmma_f32_16x16x4_f32(
          /*neg_a=*/false, a, /*neg_b=*/false, b,
          /*c_mod=*/(short)0, acc[t], /*reuse_a=*/false, /*reuse_b=*/false);
    }
  }

  // C/D layout: VGPR r -> row (row0 + r + 8*half), col = lrow
#pragma unroll
  for (int t = 0; t < Fo / 16; ++t) {
#pragma unroll
    for (int r = 0; r < 8; ++r) {
      out[(size_t)(row0 + r + 8 * half) * Fo + t * 16 + lrow] = acc[t][r];
    }
  }
}

// ------------------- y = xt * dinv^2 + b ; zero BN stats -------------------
template <int Fo>
__global__ void k_init_y(const float* __restrict__ xt,
                         const float* __restrict__ dinv,
                         const float* __restrict__ bias,
                         float* __restrict__ y,
                         float* __restrict__ stats) {
  size_t idx = (size_t)blockIdx.x * blockDim.x + threadIdx.x;
  if (idx < 2 * Fo) stats[idx] = 0.0f;
  if (idx >= (size_t)kN * Fo) return;
  int i = (int)(idx / Fo);
  int f = (int)(idx - (size_t)i * Fo);
  float di = dinv[i];
  y[idx] = xt[idx] * di * di + bias[f];
}

// ------------- edge scatter: y[dst] += xt[src] * dinv_s * dinv_d -----------
// One thread per (edge, 4-feature chunk): coalesced float4 gather + 4 atomics.
template <int Fo>
__global__ void k_edge_agg(const float* __restrict__ xt,
                           const int* __restrict__ src,
                           const int* __restrict__ dst,
                           const float* __restrict__ dinv,
                           float* __restrict__ y) {
  constexpr int C = Fo / 4;  // float4 chunks per edge
  long long idx = (long long)blockIdx.x * blockDim.x + threadIdx.x;
  if (idx >= (long long)kE * C) return;
  int e  = (int)(idx / C);
  int f4 = (int)(idx - (long long)e * C) * 4;
  int s = src[e], d = dst[e];
  float nrm = dinv[s] * dinv[d];
  float4 v = *(const float4*)(xt + (size_t)s * Fo + f4);
  float* yp = y + (size_t)d * Fo + f4;
  atomicAdd(yp + 0, v.x * nrm);
  atomicAdd(yp + 1, v.y * nrm);
  atomicAdd(yp + 2, v.z * nrm);
  atomicAdd(yp + 3, v.w * nrm);
}

// --------------- per-column sum / sumsq (LDS reduce + atomics) -------------
template <int Fo>
__global__ void k_stats(const float* __restrict__ y, float* __restrict__ stats) {
  __shared__ float sh[2 * 256];
  int tid  = threadIdx.x;                          // blockDim.x == 256
  int gt   = blockIdx.x * 256 + tid;
  int f    = gt % Fo;
  int r0   = gt / Fo;
  int rstr = (gridDim.x * 256) / Fo;               // 256*grid divisible by Fo
  float s = 0.0f, s2 = 0.0f;
  for (int r = r0; r < kN; r += rstr) {
    float v = y[(size_t)r * Fo + f];
    s += v;
    s2 += v * v;
  }
  sh[tid] = s;
  sh[256 + tid] = s2;
  __syncthreads();
  if (tid < Fo) {
    float a = 0.0f, a2 = 0.0f;
#pragma unroll
    for (int g = 0; g < 256 / Fo; ++g) {
      a  += sh[g * Fo + tid];
      a2 += sh[256 + g * Fo + tid];
    }
    atomicAdd(&stats[tid], a);
    atomicAdd(&stats[Fo + tid], a2);
  }
}

// --------------------------- BN + ReLU -------------------------------------
template <int Fo>
__global__ void k_bnrelu(const float* __restrict__ y,
                         const float* __restrict__ stats,
                         const float* __restrict__ gamma,
                         const float* __restrict__ beta,
                         float* __restrict__ hout) {
  size_t idx = (size_t)blockIdx.x * blockDim.x + threadIdx.x;
  if (idx >= (size_t)kN * Fo) return;
  int f = (int)(idx % Fo);
  const float invn = 1.0f / (float)kN;
  float m  = stats[f] * invn;
  float vv = stats[Fo + f] * invn - m * m;           // biased variance
  float sc = gamma[f] * rsqrtf(vv + kEps);
  float r  = (y[idx] - m) * sc + beta[f];
  hout[idx] = r > 0.0f ? r : 0.0f;
}

// --------------------------- pooling + head --------------------------------
__global__ void k_pool_zero(float* __restrict__ pooled, float* __restrict__ cnt) {
  int i = blockIdx.x * blockDim.x + threadIdx.x;
  if (i < kG * 32) pooled[i] = 0.0f;
  if (i < kG) cnt[i] = 0.0f;
}

__global__ void k_pool(const float* __restrict__ h, const int* __restrict__ batch,
                       float* __restrict__ pooled, float* __restrict__ cnt) {
  size_t idx = (size_t)blockIdx.x * blockDim.x + threadIdx.x;
  if (idx >= (size_t)kN * 32) return;
  int i = (int)(idx >> 5);
  int f = (int)(idx & 31);
  int g = batch[i];
  atomicAdd(&pooled[g * 32 + f], h[idx]);
  if (f == 0) atomicAdd(&cnt[g], 1.0f);
}

__global__ void k_head(const float* __restrict__ pooled, const float* __restrict__ cnt,
                       const float* __restrict__ Wf1, const float* __restrict__ bf1,
                       const float* __restrict__ Wf2, const float* __restrict__ bf2,
                       float* __restrict__ out) {
  int g = threadIdx.x;
  if (g >= kG) return;
  float inv = 1.0f / fmaxf(cnt[g], 1.0f);
  float p[32];
#pragma unroll
  for (int k = 0; k < 32; ++k) p[k] = pooled[g * 32 + k] * inv;
  float o = bf2[0];
#pragma unroll
  for (int j = 0; j < 16; ++j) {
    float a = bf1[j];
#pragma unroll
    for (int k = 0; k < 32; ++k) a += p[k] * Wf1[k * 16 + j];
    a = a > 0.0f ? a : 0.0f;
    o += a * Wf2[j];
  }
  out[g] = o;
}

// ---------------------------------------------------------------------------
extern "C" void kernel_launch(void* const* d_in, const int* in_sizes, int n_in,
                              void* d_out, int out_size, void* d_ws, size_t ws_size,
                              hipStream_t stream) {
  (void)in_sizes; (void)n_in; (void)out_size; (void)ws_size;

  const float* x    = (const float*)d_in[0];
  const int*   ei   = (const int*)d_in[1];      // [2, E]: row0 = src, row1 = dst
  const int*   bat  = (const int*)d_in[2];
  const float* W1   = (const float*)d_in[3];
  const float* b1   = (const float*)d_in[4];
  const float* W2   = (const float*)d_in[5];
  const float* b2   = (const float*)d_in[6];
  const float* W3   = (const float*)d_in[7];
  const float* b3   = (const float*)d_in[8];
  const float* g1   = (const float*)d_in[9];
  const float* be1  = (const float*)d_in[10];
  const float* g2   = (const float*)d_in[11];
  const float* be2  = (const float*)d_in[12];
  const float* g3   = (const float*)d_in[13];
  const float* be3  = (const float*)d_in[14];
  const float* Wf1  = (const float*)d_in[15];
  const float* bf1  = (const float*)d_in[16];
  const float* Wf2  = (const float*)d_in[17];
  const float* bf2  = (const float*)d_in[18];
  float* out = (float*)d_out;

  const int* src = ei;
  const int* dst = ei + kE;

  // workspace layout (floats)
  float* ws     = (float*)d_ws;
  float* deg    = ws;                               // N
  float* dinv   = deg + kN;                         // N
  float* xt     = dinv + kN;                        // N*64
  float* y      = xt + (size_t)kN * 64;             // N*64
  float* hA     = y + (size_t)kN * 64;              // N*64
  float* hB     = hA + (size_t)kN * 64;             // N*64
  float* stats  = hB + (size_t)kN * 64;             // 128
  float* pooled = stats + 128;                      // G*32
  float* cntb   = pooled + kG * 32;                 // G

  const int TB = 256;
  auto blocks = [](long long n, int tb) { return (unsigned)((n + tb - 1) / tb); };

  // degrees
  k_deg_init<<<blocks(kN, TB), TB, 0, stream>>>(deg);
  k_deg_acc<<<blocks(kE, TB), TB, 0, stream>>>(dst, deg);
  k_dinv<<<blocks(kN, TB), TB, 0, stream>>>(deg, dinv);

  const unsigned gemm_blocks = blocks(kTiles, 8);   // 8 waves (tiles) per block

  // ---- Layer 1: 128 -> 32 ----
  k_gemm_wmma<128, 32><<<gemm_blocks, 256, 0, stream>>>(x, W1, xt);
  k_init_y<32><<<blocks((long long)kN * 32, TB), TB, 0, stream>>>(xt, dinv, b1, y, stats);
  k_edge_agg<32><<<blocks((long long)kE * 8, TB), TB, 0, stream>>>(xt, src, dst, dinv, y);
  k_stats<32><<<512, 256, 0, stream>>>(y, stats);
  k_bnrelu<32><<<blocks((long long)kN * 32, TB), TB, 0, stream>>>(y, stats, g1, be1, hA);

  // ---- Layer 2: 32 -> 64 ----
  k_gemm_wmma<32, 64><<<gemm_blocks, 256, 0, stream>>>(hA, W2, xt);
  k_init_y<64><<<blocks((long long)kN * 64, TB), TB, 0, stream>>>(xt, dinv, b2, y, stats);
  k_edge_agg<64><<<blocks((long long)kE * 16, TB), TB, 0, stream>>>(xt, src, dst, dinv, y);
  k_stats<64><<<512, 256, 0, stream>>>(y, stats);
  k_bnrelu<64><<<blocks((long long)kN * 64, TB), TB, 0, stream>>>(y, stats, g2, be2, hB);

  // ---- Layer 3: 64 -> 32 ----
  k_gemm_wmma<64, 32><<<gemm_blocks, 256, 0, stream>>>(hB, W3, xt);
  k_init_y<32><<<blocks((long long)kN * 32, TB), TB, 0, stream>>>(xt, dinv, b3, y, stats);
  k_edge_agg<32><<<blocks((long long)kE * 8, TB), TB, 0, stream>>>(xt, src, dst, dinv, y);
  k_stats<32><<<512, 256, 0, stream>>>(y, stats);
  k_bnrelu<32><<<blocks((long long)kN * 32, TB), TB, 0, stream>>>(y, stats, g3, be3, hA);

  // ---- pool + head ----
  k_pool_zero<<<blocks(kG * 32, TB), TB, 0, stream>>>(pooled, cntb);
  k_pool<<<blocks((long long)kN * 32, TB), TB, 0, stream>>>(hA, bat, pooled, cntb);
  k_head<<<1, 64, 0, stream>>>(pooled, cntb, Wf1, bf1, Wf2, bf2, out);
}